// SignatureScoreLoss_42906723287572
// MI455X (gfx1250) — compile-verified
//
#include <hip/hip_runtime.h>
#include <hip/hip_bf16.h>

// SignatureScoreLoss for MI455X (gfx1250, wave32).
//
// Pipeline (big-workspace path):
//   prep_kernel     : build padded increment table E[17][256][12] (sample 16 = target)
//   mm_wmma_kernel  : M[p] = E_i * E_j^T via v_wmma_f32_16x16x4_f32 (3 WMMA / 16x16 tile)
//   pde_kernel      : per-pair Goursat wavefront, 1 base row per thread (2 refined rows),
//                     509 systolic steps, 3-phase LDS ring handoff, 1 barrier/step
//   reduce_kernel   : weighted sum of 136 clamped corner values -> scalar loss
// Fallback (small ws): pde_fused_kernel recomputes the K=9 dot per step from LDS.

#define M_SAMPLES 16
#define S_LEN     256
#define D_FEAT    8
#define NROW      255            // base increment grid
#define N_CROSS   120
#define N_PAIRS   136
#define N_STEPS   509            // 255 + 255 - 1 systolic steps
#define LAMBDA_C  0.5f
#define CLAMP_LO  (-10.0f)
#define CLAMP_HI  (10.0f)

typedef __attribute__((ext_vector_type(2))) float v2f;
typedef __attribute__((ext_vector_type(8))) float v8f;

__device__ __forceinline__ void decode_pair(int p, int& i, int& j) {
    if (p < N_CROSS) {
        int rem = p; int ii = 0;
        while (rem >= (M_SAMPLES - 1 - ii)) { rem -= (M_SAMPLES - 1 - ii); ++ii; }
        i = ii; j = ii + 1 + rem;
    } else {
        i = p - N_CROSS; j = M_SAMPLES;   // j==16 -> target path
    }
}

// ---------------------------------------------------------------------------
// Build E[sIdx][row][k], sIdx in 0..16, row in 0..255 (row 255 zero pad),
// k: 0 = time increment, 1..8 = feature increments, 9..11 = zero pad (K=12).
// ---------------------------------------------------------------------------
__global__ void prep_kernel(const float* __restrict__ gen,
                            const float* __restrict__ tgt,
                            const float* __restrict__ tp,
                            float* __restrict__ E) {
    const int sIdx = blockIdx.x;          // 0..16
    const int t    = threadIdx.x;         // 0..255
    const float* src = (sIdx < M_SAMPLES) ? (gen + (size_t)sIdx * S_LEN * D_FEAT) : tgt;
    float* row = E + ((size_t)sIdx * 256 + t) * 12;
    if (t < NROW) {
        row[0] = tp[t + 1] - tp[t];
#pragma unroll
        for (int k = 0; k < D_FEAT; ++k)
            row[1 + k] = src[(t + 1) * D_FEAT + k] - src[t * D_FEAT + k];
        row[9] = 0.0f; row[10] = 0.0f; row[11] = 0.0f;
    } else {
#pragma unroll
        for (int k = 0; k < 12; ++k) row[k] = 0.0f;
    }
}

// ---------------------------------------------------------------------------
// Batched inner-product GEMM via f32 WMMA: one block per pair, 8 waves,
// each wave owns 2 tile-rows x 16 tile-cols of the padded 256x256 output.
// A fragment (16x4 f32): lane l holds rows (l&15), K = kc*4 + 2*(l>>4) + {0,1}.
// B fragment (4x16 f32): lane l holds col  (l&15), same K striping.
// D (16x16 f32): vgpr v, lane l -> (v + 8*(l>>4), l&15).
// ---------------------------------------------------------------------------
__global__ void mm_wmma_kernel(const float* __restrict__ E,
                               float* __restrict__ M) {
    const int p    = blockIdx.x;
    const int lane = threadIdx.x & 31;
    const int wv   = threadIdx.x >> 5;          // 0..7
    const int half = lane >> 4;                  // 0 or 1
    const int lm   = lane & 15;

    int i, j; decode_pair(p, i, j);
    const float* Ei = E + (size_t)i * 256 * 12;
    const float* Ej = E + (size_t)j * 256 * 12;
    float* Mp = M + (size_t)p * 256 * 256;

#pragma unroll
    for (int trl = 0; trl < 2; ++trl) {
        const int tr = wv * 2 + trl;
        const int arow = tr * 16 + lm;
        v2f a[3];
#pragma unroll
        for (int kc = 0; kc < 3; ++kc)
            a[kc] = *(const v2f*)(Ei + (size_t)arow * 12 + kc * 4 + 2 * half);

        for (int tc = 0; tc < 16; ++tc) {
            const int bcol = tc * 16 + lm;
            v2f bf[3];
#pragma unroll
            for (int kc = 0; kc < 3; ++kc)
                bf[kc] = *(const v2f*)(Ej + (size_t)bcol * 12 + kc * 4 + 2 * half);

            v8f acc = {};
#pragma unroll
            for (int kc = 0; kc < 3; ++kc)
                acc = __builtin_amdgcn_wmma_f32_16x16x4_f32(
                        false, a[kc], false, bf[kc], (short)0, acc, false, false);

            float* dst = Mp + (size_t)(tr * 16) * 256 + tc * 16;
#pragma unroll
            for (int v = 0; v < 8; ++v)
                dst[(size_t)(v + 8 * half) * 256 + lm] = acc[v];
        }
    }
}

// ---------------------------------------------------------------------------
// Goursat wavefront core. Thread b owns base row b (refined rows 2b+1, 2b+2);
// at step s it processes base column c = s - b (refined cols 2c+1, 2c+2),
// all four refined cells sharing inc = M[b][c]/4:
//   A = 1 + inc/2 + inc^2/12,  B = 1 - inc^2/12
//   K[i,j] = (K[i,j-1] + K[i-1,j])*A - K[i-1,j-1]*B
// Handoff of the bottom refined row to thread b+1 via 3-phase LDS ring.
// ---------------------------------------------------------------------------
__device__ __forceinline__ void goursat_step(float m, float& r1, float& r2,
                                             float u0, float u1, float u2,
                                             float& bot1, float& bot2) {
    const float A  = 1.0f + 0.5f * m + m * m * (1.0f / 12.0f);
    const float Bc = 1.0f - m * m * (1.0f / 12.0f);
    const float k11 = (r1 + u1) * A - u0 * Bc;
    const float k12 = (k11 + u2) * A - u1 * Bc;
    const float k21 = (r2 + k11) * A - r1 * Bc;
    const float k22 = (k21 + k12) * A - k11 * Bc;
    r1 = k12; r2 = k22; bot1 = k21; bot2 = k22;
}

__global__ void pde_kernel(const float* __restrict__ M,
                           float* __restrict__ kvals) {
    __shared__ float2 cbuf[3][256];
    const int p = blockIdx.x;
    const int b = threadIdx.x;
    const float* Mrow = M + (size_t)p * 256 * 256 + (size_t)b * 256;

    float r1 = 1.0f, r2 = 1.0f;
    float mcur = (b == 0) ? Mrow[0] : 0.0f;   // pipeline seed for b==0 (others fill via rotate)

    for (int s = 0; s < N_STEPS; ++s) {
        const int c = s - b;
        float mnext = 0.0f;
        if (b < NROW && (c + 1) >= 0 && (c + 1) < NROW) mnext = Mrow[c + 1];  // 1-deep prefetch

        if (b < NROW && c >= 0 && c < NROW) {
            float u0, u1, u2;
            if (b == 0) { u0 = 1.0f; u1 = 1.0f; u2 = 1.0f; }
            else {
                const float2 up = cbuf[(s + 2) % 3][b - 1];      // step s-1
                u1 = up.x; u2 = up.y;
                u0 = (c == 0) ? 1.0f : cbuf[(s + 1) % 3][b - 1].y; // step s-2
            }
            float bot1, bot2;
            goursat_step(mcur * 0.25f, r1, r2, u0, u1, u2, bot1, bot2);
            cbuf[s % 3][b] = make_float2(bot1, bot2);
        }
        mcur = mnext;
        __syncthreads();
    }
    if (b == NROW - 1) {
        kvals[p] = fminf(fmaxf(r2, CLAMP_LO), CLAMP_HI);
    }
}

// Fallback: identical wavefront but recomputes inc via K=9 dot from LDS-staged
// increments (used only when workspace is too small to hold M).
__global__ void pde_fused_kernel(const float* __restrict__ gen,
                                 const float* __restrict__ tgt,
                                 const float* __restrict__ tp,
                                 float* __restrict__ kvals) {
    __shared__ float  Ejs[NROW * 12];
    __shared__ float2 cbuf[3][256];
    const int p = blockIdx.x;
    const int b = threadIdx.x;

    int i, j; decode_pair(p, i, j);
    const float* srcJ = (j < M_SAMPLES) ? (gen + (size_t)j * S_LEN * D_FEAT) : tgt;
    if (b < NROW) {
        float* r = &Ejs[b * 12];
        r[0] = tp[b + 1] - tp[b];
#pragma unroll
        for (int k = 0; k < D_FEAT; ++k)
            r[1 + k] = srcJ[(b + 1) * D_FEAT + k] - srcJ[b * D_FEAT + k];
        r[9] = 0.0f; r[10] = 0.0f; r[11] = 0.0f;
    }
    float ei[12];
#pragma unroll
    for (int k = 0; k < 12; ++k) ei[k] = 0.0f;
    if (b < NROW) {
        const float* srcI = gen + (size_t)i * S_LEN * D_FEAT;
        ei[0] = tp[b + 1] - tp[b];
#pragma unroll
        for (int k = 0; k < D_FEAT; ++k)
            ei[1 + k] = srcI[(b + 1) * D_FEAT + k] - srcI[b * D_FEAT + k];
    }
    __syncthreads();

    float r1 = 1.0f, r2 = 1.0f;
    for (int s = 0; s < N_STEPS; ++s) {
        const int c = s - b;
        if (b < NROW && c >= 0 && c < NROW) {
            const float* ej = &Ejs[c * 12];
            float m = 0.0f;
#pragma unroll
            for (int k = 0; k < 12; ++k) m = fmaf(ei[k], ej[k], m);

            float u0, u1, u2;
            if (b == 0) { u0 = 1.0f; u1 = 1.0f; u2 = 1.0f; }
            else {
                const float2 up = cbuf[(s + 2) % 3][b - 1];
                u1 = up.x; u2 = up.y;
                u0 = (c == 0) ? 1.0f : cbuf[(s + 1) % 3][b - 1].y;
            }
            float bot1, bot2;
            goursat_step(m * 0.25f, r1, r2, u0, u1, u2, bot1, bot2);
            cbuf[s % 3][b] = make_float2(bot1, bot2);
        }
        __syncthreads();
    }
    if (b == NROW - 1) {
        kvals[p] = fminf(fmaxf(r2, CLAMP_LO), CLAMP_HI);
    }
}

// ---------------------------------------------------------------------------
// loss = LAMBDA * sum_cross / (m*(m-1)) - sum_tgt / m  (fixed-order tree sum)
// ---------------------------------------------------------------------------
__global__ void reduce_kernel(const float* __restrict__ kvals,
                              float* __restrict__ out) {
    __shared__ float sb[256];
    const int t = threadIdx.x;
    float v = 0.0f;
    if (t < N_PAIRS) {
        const float coef = (t < N_CROSS)
            ? (LAMBDA_C / (float)(M_SAMPLES * (M_SAMPLES - 1)))
            : (-1.0f / (float)M_SAMPLES);
        v = coef * kvals[t];
    }
    sb[t] = v;
    __syncthreads();
    for (int o = 128; o > 0; o >>= 1) {
        if (t < o) sb[t] += sb[t + o];
        __syncthreads();
    }
    if (t == 0) out[0] = sb[0];
}

extern "C" void kernel_launch(void* const* d_in, const int* in_sizes, int n_in,
                              void* d_out, int out_size, void* d_ws, size_t ws_size,
                              hipStream_t stream) {
    const float* gen = (const float*)d_in[0];   // [16,256,8]
    const float* tgt = (const float*)d_in[1];   // [256,8]
    const float* tp  = (const float*)d_in[2];   // [256]
    float* out = (float*)d_out;

    char* ws = (char*)d_ws;
    float* kvals = (float*)ws;                              // 136 floats (1 KB reserved)
    float* E     = (float*)(ws + 1024);                     // 17*256*12*4 = 208,896 B
    const size_t M_off  = (size_t)1 << 20;
    float* M     = (float*)(ws + M_off);                    // 136*256*256*4 = 35,651,584 B
    const size_t need   = M_off + (size_t)N_PAIRS * 256 * 256 * sizeof(float);

    if (ws_size >= need) {
        prep_kernel<<<M_SAMPLES + 1, 256, 0, stream>>>(gen, tgt, tp, E);
        mm_wmma_kernel<<<N_PAIRS, 256, 0, stream>>>(E, M);
        pde_kernel<<<N_PAIRS, 256, 0, stream>>>(M, kvals);
    } else {
        pde_fused_kernel<<<N_PAIRS, 256, 0, stream>>>(gen, tgt, tp, kvals);
    }
    reduce_kernel<<<1, 256, 0, stream>>>(kvals, out);
}